// WeightedGATConvGraphGymLayer_11690900979886
// MI455X (gfx1250) — compile-verified
//
#include <hip/hip_runtime.h>
#include <hip/hip_bf16.h>

typedef float v2f __attribute__((ext_vector_type(2)));
typedef float v8f __attribute__((ext_vector_type(8)));

#define HEADS 4
#define CH    32
#define DIN_  128
#define NEG_SLOPE_ 0.2f

// ---- monotone float<->uint mapping for atomicMax on floats (handles negatives) ----
__device__ __forceinline__ unsigned f2ord(float f) {
    unsigned u = __float_as_uint(f);
    return (u & 0x80000000u) ? ~u : (u | 0x80000000u);
}
__device__ __forceinline__ float ord2f(unsigned u) {
    return (u & 0x80000000u) ? __uint_as_float(u & 0x7FFFFFFFu) : __uint_as_float(~u);
}
#define ORD_NEG_INF 0x007FFFFFu  // f2ord(-inf)

__device__ __forceinline__ float leaky(float v) {
    return v > 0.0f ? v : NEG_SLOPE_ * v;
}

// ---------------- init: zero out/deg/wsum/denom, amax = -inf ----------------
__global__ void init_kernel(float* __restrict__ out, float* __restrict__ deg,
                            float* __restrict__ wsum, float* __restrict__ denom,
                            unsigned* __restrict__ amax, int n) {
    long long i = (long long)blockIdx.x * blockDim.x + threadIdx.x;
    long long tot = (long long)n * DIN_;
    if (i < tot) out[i] = 0.0f;
    if (i < n) { deg[i] = 0.0f; wsum[i] = 0.0f; }
    if (i < (long long)n * HEADS) { denom[i] = 0.0f; amax[i] = ORD_NEG_INF; }
}

// ---------------- h = x @ W via V_WMMA_F32_16X16X4_F32 ----------------
// Block = 256 threads = 8 waves. Wave w computes the 16x16 tile:
//   rows [16*blockIdx.x, +16), cols [16*w, +16) of h (DIN_=HC=128 cols).
// A (16x4 f32) lane layout: row = lane%16; vgpr0/1 hold K = 2*(lane/16) + {0,1}.
// B (4x16 f32)  lane layout: col = lane%16; vgpr0/1 hold K = 2*(lane/16) + {0,1}.
// C/D (16x16 f32): vgpr r -> row r + 8*(lane/16), col = lane%16.
__global__ void gemm_wmma_kernel(const float* __restrict__ x,
                                 const float* __restrict__ W,
                                 float* __restrict__ h, int n) {
    const int wave = threadIdx.x >> 5;
    const int lane = threadIdx.x & 31;
    const int half = lane >> 4;   // 0 or 1
    const int l16  = lane & 15;
    const int row0 = blockIdx.x * 16;
    const int col0 = wave * 16;

    int arow = row0 + l16;
    if (arow >= n) arow = n - 1;          // clamp loads for ragged tail
    const float* __restrict__ xrow = x + (size_t)arow * DIN_;
    const int kk = half * 2;

    v8f acc = {};
    #pragma unroll
    for (int k = 0; k < DIN_; k += 4) {
        v2f a, b;
        a.x = xrow[k + kk + 0];
        a.y = xrow[k + kk + 1];
        b.x = W[(size_t)(k + kk + 0) * DIN_ + col0 + l16];
        b.y = W[(size_t)(k + kk + 1) * DIN_ + col0 + l16];
        acc = __builtin_amdgcn_wmma_f32_16x16x4_f32(
            /*neg_a=*/false, a, /*neg_b=*/false, b,
            /*c_mod=*/(short)0, acc, /*reuse_a=*/false, /*reuse_b=*/false);
    }

    float* __restrict__ hp = h + (size_t)(col0 + l16);
    #pragma unroll
    for (int r = 0; r < 8; ++r) {
        int row = row0 + r + half * 8;
        if (row < n) hp[(size_t)row * DIN_] = acc[r];
    }
}

// ---------------- a_src/a_dst: per (node, head) dot with att vectors ----------------
__global__ void att_kernel(const float* __restrict__ h,
                           const float* __restrict__ att_src,
                           const float* __restrict__ att_dst,
                           float* __restrict__ a_src, float* __restrict__ a_dst, int n) {
    long long i = (long long)blockIdx.x * blockDim.x + threadIdx.x;
    if (i >= (long long)n * HEADS) return;
    const int head = (int)(i & (HEADS - 1));
    const float* __restrict__ hp = h + (i >> 2) * DIN_ + head * CH;
    const float* __restrict__ as = att_src + head * CH;
    const float* __restrict__ ad = att_dst + head * CH;
    float s = 0.0f, d = 0.0f;
    #pragma unroll
    for (int c = 0; c < CH; ++c) {
        float v = hp[c];
        s = fmaf(v, as[c], s);
        d = fmaf(v, ad[c], d);
    }
    a_src[i] = s;
    a_dst[i] = d;
}

// ---------------- edge pass 1: deg, weight sum, and edge-logit segment max ----------------
__global__ void edge_pass1_kernel(const int* __restrict__ src, const int* __restrict__ dst,
                                  const float* __restrict__ ea,
                                  const float* __restrict__ a_src, const float* __restrict__ a_dst,
                                  float* __restrict__ deg, float* __restrict__ wsum,
                                  unsigned* __restrict__ amax, int e) {
    long long i = (long long)blockIdx.x * blockDim.x + threadIdx.x;
    if (i >= e) return;
    const int s = src[i], d = dst[i];
    const float w = ea[i];
    atomicAdd(&deg[d], 1.0f);
    atomicAdd(&wsum[d], w);
    const float l2w = __log2f(w);
    #pragma unroll
    for (int hh = 0; hh < HEADS; ++hh) {
        float v = leaky(a_src[s * HEADS + hh] + a_dst[d * HEADS + hh]) + l2w;
        atomicMax(&amax[d * HEADS + hh], f2ord(v));
    }
}

// ---------------- node pass: loop_w and self-loop logit into segment max ----------------
__global__ void node_pass_kernel(const float* __restrict__ deg, const float* __restrict__ wsum,
                                 float* __restrict__ loopw,
                                 const float* __restrict__ a_src, const float* __restrict__ a_dst,
                                 unsigned* __restrict__ amax, int n) {
    long long i = (long long)blockIdx.x * blockDim.x + threadIdx.x;
    if (i >= n) return;
    const float dg = deg[i];
    const float lw = dg > 0.0f ? wsum[i] / dg : 1.0f;
    loopw[i] = lw;
    const float l2w = __log2f(lw);
    #pragma unroll
    for (int hh = 0; hh < HEADS; ++hh) {
        float v = leaky(a_src[i * HEADS + hh] + a_dst[i * HEADS + hh]) + l2w;
        atomicMax(&amax[i * HEADS + hh], f2ord(v));
    }
}

// ---------------- edge scatter: one wave per edge, 4 cols/lane ----------------
// out[dst, :] += ex * h[src, :]   (unnormalized); denom[dst, head] += ex
__global__ void scatter_kernel(const int* __restrict__ src, const int* __restrict__ dst,
                               const float* __restrict__ ea,
                               const float* __restrict__ a_src, const float* __restrict__ a_dst,
                               const unsigned* __restrict__ amax,
                               const float* __restrict__ h,
                               float* __restrict__ denom, float* __restrict__ out, int e) {
    long long gid = (long long)blockIdx.x * blockDim.x + threadIdx.x;
    long long edge = gid >> 5;
    const int lane = (int)(gid & 31);
    if (edge >= e) return;
    const int s = src[edge], d = dst[edge];
    const int head = lane >> 3;                         // cols lane*4.. -> head = lane/8
    float v = leaky(a_src[s * HEADS + head] + a_dst[d * HEADS + head]);
    v += __log2f(ea[edge]);
    const float ex = __expf(v - ord2f(amax[d * HEADS + head]));
    if ((lane & 7) == 0) atomicAdd(&denom[d * HEADS + head], ex);
    const float* __restrict__ hp = h + (size_t)s * DIN_ + lane * 4;
    float* __restrict__ op = out + (size_t)d * DIN_ + lane * 4;
    #pragma unroll
    for (int c = 0; c < 4; ++c) atomicAdd(&op[c], ex * hp[c]);
}

// ---------------- finalize: add self loop, normalize, add bias ----------------
__global__ void finalize_kernel(const float* __restrict__ h,
                                const float* __restrict__ a_src, const float* __restrict__ a_dst,
                                const float* __restrict__ loopw, const unsigned* __restrict__ amax,
                                const float* __restrict__ denom, const float* __restrict__ bias,
                                float* __restrict__ out, int n) {
    long long i = (long long)blockIdx.x * blockDim.x + threadIdx.x;
    if (i >= (long long)n * DIN_) return;
    const int node = (int)(i >> 7);
    const int col  = (int)(i & (DIN_ - 1));
    const int head = col >> 5;
    float v = leaky(a_src[node * HEADS + head] + a_dst[node * HEADS + head]);
    v += __log2f(loopw[node]);
    const float ex = __expf(v - ord2f(amax[node * HEADS + head]));
    const float den = denom[node * HEADS + head] + ex;
    out[i] = (out[i] + ex * h[i]) / den + bias[col];
}

extern "C" void kernel_launch(void* const* d_in, const int* in_sizes, int n_in,
                              void* d_out, int out_size, void* d_ws, size_t ws_size,
                              hipStream_t stream) {
    const float* x        = (const float*)d_in[0];
    const int*   eidx     = (const int*)d_in[1];
    const float* eattr    = (const float*)d_in[2];
    const float* W        = (const float*)d_in[3];
    const float* att_src  = (const float*)d_in[4];
    const float* att_dst  = (const float*)d_in[5];
    const float* bias     = (const float*)d_in[6];

    const int n = in_sizes[0] / DIN_;       // nodes
    const int e = in_sizes[2];              // edges
    const int* src = eidx;
    const int* dst = eidx + e;

    float* out = (float*)d_out;

    // workspace carve
    float*    h     = (float*)d_ws;
    float*    asrc  = h     + (size_t)n * DIN_;
    float*    adst  = asrc  + (size_t)n * HEADS;
    float*    deg   = adst  + (size_t)n * HEADS;
    float*    wsum  = deg   + n;
    float*    loopw = wsum  + n;
    float*    denom = loopw + n;
    unsigned* amax  = (unsigned*)(denom + (size_t)n * HEADS);

    const int B = 256;
    const long long nTot = (long long)n * DIN_;

    init_kernel<<<(unsigned)((nTot + B - 1) / B), B, 0, stream>>>(out, deg, wsum, denom, amax, n);

    gemm_wmma_kernel<<<(unsigned)((n + 15) / 16), B, 0, stream>>>(x, W, h, n);

    att_kernel<<<(unsigned)(((long long)n * HEADS + B - 1) / B), B, 0, stream>>>(
        h, att_src, att_dst, asrc, adst, n);

    edge_pass1_kernel<<<(unsigned)(((long long)e + B - 1) / B), B, 0, stream>>>(
        src, dst, eattr, asrc, adst, deg, wsum, amax, e);

    node_pass_kernel<<<(unsigned)((n + B - 1) / B), B, 0, stream>>>(
        deg, wsum, loopw, asrc, adst, amax, n);

    scatter_kernel<<<(unsigned)(((long long)e * 32 + B - 1) / B), B, 0, stream>>>(
        src, dst, eattr, asrc, adst, amax, h, denom, out, e);

    finalize_kernel<<<(unsigned)((nTot + B - 1) / B), B, 0, stream>>>(
        h, asrc, adst, loopw, amax, denom, bias, out, n);
}